// AdaptiveHyperNN_29703993819986
// MI455X (gfx1250) — compile-verified
//
#include <hip/hip_runtime.h>
#include <hip/hip_bf16.h>
#include <math.h>
#include <stdint.h>

typedef __attribute__((ext_vector_type(2))) float v2f;
typedef __attribute__((ext_vector_type(8))) float v8f;

#define DC 256           // channels
#define BG 8             // graphs
#define NN 128           // nodes per graph
#define MROWS (BG * NN)  // 1024 node rows
#define LDSROW 260       // padded LDS row stride (floats): 1040B rows -> 16B aligned,
                         // bank step 4 across the 16 A-rows (no conflicts)

// ---------------------------------------------------------------------------
// Stage 1 (fused): blocks 0..7  : gather feat, per-graph mean (in LDS), then
//                                 g[b,j] = featmean[b]·W1_top[:,j] + b1[j]
//                  block  8     : p = W3_top@W4_top, q = W3_bot@W4_top
//                  block  9     : d[b] = b3·W4_top + Xs[b]·W4_bot + b4
__global__ __launch_bounds__(DC)
void k_prep(const float* __restrict__ api_embeds, const int* __restrict__ invoked,
            const float* __restrict__ W1, const float* __restrict__ b1,
            const float* __restrict__ W3, const float* __restrict__ W4,
            const float* __restrict__ b3, const float* __restrict__ Xs,
            const float* __restrict__ b4,
            float* __restrict__ feat, float* __restrict__ g,
            float* __restrict__ p, float* __restrict__ q,
            float* __restrict__ dvec) {
  const int t = threadIdx.x;
  const int bid = blockIdx.x;
  if (bid < BG) {
    __shared__ float sm[DC];
    float s = 0.f;
    for (int n = 0; n < NN; ++n) {
      const int api = invoked[bid * NN + n];
      const float v = api_embeds[api * DC + t];
      feat[(bid * NN + n) * DC + t] = v;
      s += v;
    }
    sm[t] = s * (1.0f / 128.0f);
    __syncthreads();
    float acc = b1[t];
    for (int k = 0; k < DC; ++k) acc = fmaf(sm[k], W1[k * DC + t], acc);
    g[bid * DC + t] = acc;
  } else if (bid == BG) {
    float sp = 0.f, sq = 0.f;
    for (int j = 0; j < DC; ++j) {
      const float w = W4[j];
      sp = fmaf(W3[t * DC + j], w, sp);
      sq = fmaf(W3[(DC + t) * DC + j], w, sq);
    }
    p[t] = sp;
    q[t] = sq;
  } else if (t < BG) {
    float s = b4[0];
    for (int j = 0; j < DC; ++j) s = fmaf(b3[j], W4[j], s);
    for (int k = 0; k < DC; ++k) s = fmaf(Xs[t * DC + k], W4[DC + k], s);
    dvec[t] = s;
  }
}

// ---------------------------------------------------------------------------
// Async-stage a 16x256 f32 panel (16KB, contiguous rows of 1KB) from global
// into LDS with padded row stride LDSROW, using CDNA5 async-to-LDS DMA.
// 1024 x 16B chunks across 256 threads (4 issues each), tracked by ASYNCcnt.
__device__ __forceinline__ void async_stage_panel(const float* __restrict__ gpanel,
                                                  float* __restrict__ lds_base,
                                                  int tid) {
  const unsigned ldsb = (unsigned)(uintptr_t)lds_base;  // LDS byte offset (low 32 bits)
#pragma unroll
  for (int i = 0; i < 4; ++i) {
    const int c = tid + 256 * i;  // chunk index, 16B each
    const int row = c >> 6;       // 64 chunks per 1KB row
    const int col = c & 63;
    const unsigned lds_addr = ldsb + (unsigned)(row * (LDSROW * 4) + col * 16);
    const unsigned goff = (unsigned)(row * 1024 + col * 16);  // byte offset in panel
    asm volatile("global_load_async_to_lds_b128 %0, %1, %2"
                 :: "v"(lds_addr), "v"(goff), "s"(gpanel)
                 : "memory");
  }
}

__device__ __forceinline__ void wait_async_all() {
  asm volatile("s_wait_asynccnt 0x0" ::: "memory");
}

// WMMA f32 16x16x4 K=256 accumulation.
//   A 16x4:  lane&15 = M;  vgpr pair = K {kh, kh+1}, kh = 2*(lane>=16)
//   B 4x16:  lane&15 = N;  K = vgpr + 2*(lane>=16)
//   C 16x16: lane&15 = N;  M = vgpr + 8*(lane>=16)
// A comes from the LDS panel (row stride LDSROW), B from global (row stride DC).
__device__ __forceinline__ v8f wmma_acc_k256_lds(const float* __restrict__ Arow,
                                                 const float* __restrict__ Bcol,
                                                 int kh, v8f acc) {
#pragma unroll 4
  for (int k = 0; k < DC; k += 4) {
    v2f a, b;
    a.x = Arow[k + kh];
    a.y = Arow[k + kh + 1];
    b.x = Bcol[(k + kh) * DC];
    b.y = Bcol[(k + kh + 1) * DC];
    acc = __builtin_amdgcn_wmma_f32_16x16x4_f32(
        /*neg_a=*/false, a, /*neg_b=*/false, b,
        /*c_mod=*/(short)0, acc, /*reuse_a=*/false, /*reuse_b=*/false);
  }
  return acc;
}

// ---------------------------------------------------------------------------
// Stage 2: e2n1 = feat @ W1_bot + g[b]   ([1024x256] x [256x256])
// Block = 8 waves sharing one async-staged 16-row A panel; wave w -> N-tile.
__global__ __launch_bounds__(256)
void k_gemm1(const float* __restrict__ feat, const float* __restrict__ W1,
             const float* __restrict__ g, float* __restrict__ e2n1) {
  __shared__ float sA[16 * LDSROW];
  const int tid = threadIdx.x;
  const int lane = tid & 31;
  const int w = tid >> 5;
  const int mt = blockIdx.x >> 1;              // 0..63
  const int nt = (blockIdx.x & 1) * 8 + w;     // 0..15
  const int kh = (lane >> 4) << 1;
  const int nc = nt * 16 + (lane & 15);

  async_stage_panel(feat + mt * 16 * DC, sA, tid);
  wait_async_all();
  __syncthreads();

  const float* Arow = sA + (lane & 15) * LDSROW;
  const float* Bcol = W1 + DC * DC + nc;       // bottom half of W1
  v8f acc = {};
  acc = wmma_acc_k256_lds(Arow, Bcol, kh, acc);

  const int gb = mt >> 3;                      // graph index (16 rows in one graph)
  const int rbase = mt * 16 + ((lane >> 4) << 3);
  const float gadd = g[gb * DC + nc];
#pragma unroll
  for (int i = 0; i < 8; ++i)
    e2n1[(rbase + i) * DC + nc] = acc[i] + gadd;
}

// Stage 3: h = feat @ W2_top + e2n1 @ W2_bot + b2  (K=512 as two K=256 passes,
// both A panels async-staged into LDS)
__global__ __launch_bounds__(256)
void k_gemm2(const float* __restrict__ feat, const float* __restrict__ e2n1,
             const float* __restrict__ W2, const float* __restrict__ b2,
             float* __restrict__ h) {
  __shared__ float sA1[16 * LDSROW];
  __shared__ float sA2[16 * LDSROW];
  const int tid = threadIdx.x;
  const int lane = tid & 31;
  const int w = tid >> 5;
  const int mt = blockIdx.x >> 1;
  const int nt = (blockIdx.x & 1) * 8 + w;
  const int kh = (lane >> 4) << 1;
  const int nc = nt * 16 + (lane & 15);

  async_stage_panel(feat + mt * 16 * DC, sA1, tid);
  async_stage_panel(e2n1 + mt * 16 * DC, sA2, tid);
  wait_async_all();
  __syncthreads();

  const float* A1 = sA1 + (lane & 15) * LDSROW;
  const float* A2 = sA2 + (lane & 15) * LDSROW;
  const float* B1 = W2 + nc;                   // W2 rows 0..255
  const float* B2 = W2 + DC * DC + nc;         // W2 rows 256..511
  v8f acc = {};
  acc = wmma_acc_k256_lds(A1, B1, kh, acc);
  acc = wmma_acc_k256_lds(A2, B2, kh, acc);

  const int rbase = mt * 16 + ((lane >> 4) << 3);
  const float badd = b2[nc];
#pragma unroll
  for (int i = 0; i < 8; ++i)
    h[(rbase + i) * DC + nc] = acc[i] + badd;
}

// ---------------------------------------------------------------------------
// Stage 4 (fused): per graph b, compute a[u]=h[b,u]·p, c[v]=h[b,v]·q with
// wave32 shuffle reductions, then write all 16384 sigmoid edge outputs.
__global__ __launch_bounds__(256)
void k_edge(const float* __restrict__ h, const float* __restrict__ p,
            const float* __restrict__ q, const float* __restrict__ dvec,
            float* __restrict__ out) {
  __shared__ float sa[NN];
  __shared__ float sc[NN];
  __shared__ float sp[DC];
  __shared__ float sq[DC];
  const int b = blockIdx.x;
  const int t = threadIdx.x;
  const int lane = t & 31;
  const int w = t >> 5;
  sp[t] = p[t];
  sq[t] = q[t];
  __syncthreads();
  // 8 waves x 16 rows; each dot split 8 cols/lane + 5-step xor reduce (wave32)
  for (int i = 0; i < 16; ++i) {
    const int r = w * 16 + i;
    const float* hr = h + (b * NN + r) * DC;
    float pa = 0.f, pc = 0.f;
#pragma unroll
    for (int j = 0; j < 8; ++j) {
      const int col = lane + 32 * j;
      const float hv = hr[col];
      pa = fmaf(hv, sp[col], pa);
      pc = fmaf(hv, sq[col], pc);
    }
#pragma unroll
    for (int m = 16; m >= 1; m >>= 1) {
      pa += __shfl_xor(pa, m);
      pc += __shfl_xor(pc, m);
    }
    if (lane == 0) { sa[r] = pa; sc[r] = pc; }
  }
  __syncthreads();
  const float db = dvec[b];
#pragma unroll 4
  for (int j = 0; j < 64; ++j) {
    const int idx = j * 256 + t;       // 0..16383
    const int u = idx >> 7;
    const int v = idx & 127;
    const float z = sa[u] + sc[v] + db;
    out[b * (NN * NN) + idx] = 1.0f / (1.0f + __expf(-z));
  }
}

// ---------------------------------------------------------------------------
extern "C" void kernel_launch(void* const* d_in, const int* in_sizes, int n_in,
                              void* d_out, int out_size, void* d_ws, size_t ws_size,
                              hipStream_t stream) {
  const float* Xs  = (const float*)d_in[0];
  const float* api = (const float*)d_in[1];
  const float* W1  = (const float*)d_in[2];
  const float* b1  = (const float*)d_in[3];
  const float* W2  = (const float*)d_in[4];
  const float* b2  = (const float*)d_in[5];
  const float* W3  = (const float*)d_in[6];
  const float* b3  = (const float*)d_in[7];
  const float* W4  = (const float*)d_in[8];
  const float* b4  = (const float*)d_in[9];
  const int* invoked = (const int*)d_in[10];

  float* ws = (float*)d_ws;
  float* feat = ws;                   // 1024*256
  float* e2n1 = feat + MROWS * DC;    // 1024*256
  float* h    = e2n1 + MROWS * DC;    // 1024*256
  float* g    = h + MROWS * DC;       // 8*256
  float* p    = g + BG * DC;          // 256
  float* q    = p + DC;               // 256
  float* dvec = q + DC;               // 8
  // ~3.02 MB f32 workspace

  k_prep<<<BG + 2, DC, 0, stream>>>(api, invoked, W1, b1, W3, W4, b3, Xs, b4,
                                    feat, g, p, q, dvec);
  k_gemm1<<<(MROWS / 16) * 2, 256, 0, stream>>>(feat, W1, g, e2n1);
  k_gemm2<<<(MROWS / 16) * 2, 256, 0, stream>>>(feat, e2n1, W2, b2, h);
  k_edge<<<BG, 256, 0, stream>>>(h, p, q, dvec, (float*)d_out);
}